// GraphAttentionNetwork_28750511079436
// MI455X (gfx1250) — compile-verified
//
#include <hip/hip_runtime.h>
#include <hip/hip_bf16.h>
#include <math.h>

#define N_NODES 10000
#define N_EDGES 320000
#define IN_F 256
#define OUT_F 256
#define SLOPE 0.2f

typedef __attribute__((ext_vector_type(2))) float v2f;
typedef __attribute__((ext_vector_type(8))) float v8f;

// ---------------------------------------------------------------------------
// 1) Wh = features @ W + b_lin   via V_WMMA_F32_16X16X4_F32 (full f32 path)
//    grid = (625, 4), block = (32, 4): each wave owns one 16x16 output tile.
// ---------------------------------------------------------------------------
__global__ void __launch_bounds__(128)
gat_wh_gemm(const float* __restrict__ A,      // [N_NODES, IN_F]
            const float* __restrict__ B,      // [IN_F, OUT_F]
            const float* __restrict__ bias,   // [OUT_F]
            float* __restrict__ Wh)           // [N_NODES, OUT_F]
{
    const int tileM = blockIdx.x;                       // 0..624
    const int tileN = blockIdx.y * 4 + threadIdx.y;     // 0..15
    const int lane  = threadIdx.x;                      // 0..31 (wave32)
    const int l     = lane & 15;
    const int h     = lane >> 4;

    const int row = tileM * 16 + l;   // A: M index held per-lane (lanes 0-15 / 16-31 both M=l)
    const int col = tileN * 16 + l;   // B/C/D: N index held per-lane

    const float* __restrict__ arow = A + (size_t)row * IN_F;

    v8f acc = {};
    for (int k = 0; k < IN_F; k += 4) {
        const int ka = k + 2 * h;               // half selects K pair {0,1} vs {2,3}
        v2f a = *(const v2f*)(arow + ka);       // A VGPR0=K(ka), VGPR1=K(ka+1)
        v2f b;
        b.x = B[(size_t)ka * OUT_F + col];      // B VGPR0 = row K=ka
        b.y = B[(size_t)(ka + 1) * OUT_F + col];
        acc = __builtin_amdgcn_wmma_f32_16x16x4_f32(
            /*neg_a=*/false, a, /*neg_b=*/false, b,
            /*c_mod=*/(short)0, acc, /*reuse_a=*/false, /*reuse_b=*/false);
    }

    const float bl = bias[col];
    const int mBase = tileM * 16 + h * 8;       // C/D: lanes 16-31 hold M = 8 + r
#pragma unroll
    for (int r = 0; r < 8; ++r)
        Wh[(size_t)(mBase + r) * OUT_F + col] = acc[r] + bl;
}

// ---------------------------------------------------------------------------
// 2) Per-node attention scores: s_src[i]=Wh[i]·Wa_src, s_dst[i]=Wh[i]·Wa_dst
//    One wave per node; wave32 xor-shuffle reduction.
// ---------------------------------------------------------------------------
__global__ void __launch_bounds__(256)
gat_scores(const float* __restrict__ Wh,
           const float* __restrict__ wa_s,
           const float* __restrict__ wa_d,
           float* __restrict__ ssrc,
           float* __restrict__ sdst)
{
    const int wave = threadIdx.x >> 5;
    const int lane = threadIdx.x & 31;
    const int node = blockIdx.x * 8 + wave;
    if (node >= N_NODES) return;

    const float* __restrict__ row = Wh + (size_t)node * OUT_F;
    float a = 0.f, b = 0.f;
    for (int c = lane; c < OUT_F; c += 32) {
        const float x = row[c];
        a += x * wa_s[c];
        b += x * wa_d[c];
    }
#pragma unroll
    for (int m = 16; m > 0; m >>= 1) {
        a += __shfl_xor(a, m, 32);
        b += __shfl_xor(b, m, 32);
    }
    if (lane == 0) { ssrc[node] = a; sdst[node] = b; }
}

// ---------------------------------------------------------------------------
// 3) colsum = sum_j Wh[j]  (deterministic two-stage reduction)
// ---------------------------------------------------------------------------
__global__ void __launch_bounds__(256)
gat_colsum_partial(const float* __restrict__ Wh, float* __restrict__ part)
{
    const int b = blockIdx.x;    // 0..63
    const int f = threadIdx.x;   // 0..255
    float s = 0.f;
    for (int r = b; r < N_NODES; r += 64)
        s += Wh[(size_t)r * OUT_F + f];
    part[b * OUT_F + f] = s;
}

__global__ void __launch_bounds__(256)
gat_colsum_final(const float* __restrict__ part, float* __restrict__ colsum)
{
    const int f = threadIdx.x;
    float s = 0.f;
    for (int b = 0; b < 64; ++b)
        s += part[b * OUT_F + f];
    colsum[f] = s;
}

// ---------------------------------------------------------------------------
// 4) Symmetric CSR construction: degree count -> scan -> fill
// ---------------------------------------------------------------------------
__global__ void __launch_bounds__(256)
gat_degree(const int* __restrict__ eu, const int* __restrict__ ev,
           int* __restrict__ deg)
{
    const int k = blockIdx.x * blockDim.x + threadIdx.x;
    if (k >= N_EDGES) return;
    const int u = eu[k], v = ev[k];
    atomicAdd(&deg[u], 1);
    if (u != v) atomicAdd(&deg[v], 1);   // self-loop: single cell in reference
}

__global__ void __launch_bounds__(1024)
gat_scan(const int* __restrict__ deg, int* __restrict__ rowptr)
{
    __shared__ int part[1024];
    const int t = threadIdx.x;
    const int base = t * 10;             // 1000 chunks of 10 cover 10000 nodes
    int s = 0;
    if (base < N_NODES) {
        for (int i = 0; i < 10; ++i) s += deg[base + i];
    }
    part[t] = s;
    __syncthreads();
#pragma unroll
    for (int off = 1; off < 1024; off <<= 1) {
        const int v = (t >= off) ? part[t - off] : 0;
        __syncthreads();
        part[t] += v;
        __syncthreads();
    }
    const int excl = (t == 0) ? 0 : part[t - 1];
    if (base < N_NODES) {
        int run = excl;
        for (int i = 0; i < 10; ++i) {
            rowptr[base + i] = run;
            run += deg[base + i];
        }
    }
    if (t == 1023) rowptr[N_NODES] = part[1023];
}

__global__ void __launch_bounds__(256)
gat_fill(const int* __restrict__ eu, const int* __restrict__ ev,
         const float* __restrict__ ssrc, const float* __restrict__ sdst,
         const float* __restrict__ batt,
         int* __restrict__ cursor, int* __restrict__ colidx,
         float* __restrict__ wval)
{
    const int k = blockIdx.x * blockDim.x + threadIdx.x;
    if (k >= N_EDGES) return;
    const int u = eu[k], v = ev[k];
    const float e  = ssrc[u] + sdst[v] + batt[0];
    const float lr = (e > 0.f) ? e : SLOPE * e;
    const float w  = __expf(lr) - 1.f;   // contribution relative to exp(0)=1 baseline
    int p = atomicAdd(&cursor[u], 1);
    colidx[p] = v; wval[p] = w;
    if (u != v) {
        int q = atomicAdd(&cursor[v], 1);
        colidx[q] = u; wval[q] = w;      // symmetric scatter uses the SAME e(u,v)
    }
}

// ---------------------------------------------------------------------------
// 5) out_i = elu( (colsum + sum_adj w*Wh_j) / (N + sum_adj w) )
//    One block per node; thread f owns feature f (coalesced Wh row reads, L2-hot).
// ---------------------------------------------------------------------------
__global__ void __launch_bounds__(256)
gat_aggregate(const float* __restrict__ Wh,
              const float* __restrict__ colsum,
              const int* __restrict__ rowptr,
              const int* __restrict__ colidx,
              const float* __restrict__ wval,
              float* __restrict__ out)
{
    const int i = blockIdx.x;
    const int f = threadIdx.x;
    float acc  = colsum[f];
    float wsum = 0.f;
    const int s = rowptr[i], e = rowptr[i + 1];
    for (int p = s; p < e; ++p) {
        const int   j = colidx[p];
        const float w = wval[p];
        wsum += w;
        acc  += w * Wh[(size_t)j * OUT_F + f];
    }
    const float x = acc / ((float)N_NODES + wsum);
    out[(size_t)i * OUT_F + f] = (x > 0.f) ? x : (__expf(x) - 1.f);
}

// ---------------------------------------------------------------------------
// Launch
// ---------------------------------------------------------------------------
extern "C" void kernel_launch(void* const* d_in, const int* in_sizes, int n_in,
                              void* d_out, int out_size, void* d_ws, size_t ws_size,
                              hipStream_t stream) {
    const float* features = (const float*)d_in[0];
    const int*   edge_u   = (const int*)  d_in[1];
    const int*   edge_v   = (const int*)  d_in[2];
    const float* W        = (const float*)d_in[3];
    const float* b_lin    = (const float*)d_in[4];
    const float* Wa_src   = (const float*)d_in[5];
    const float* Wa_dst   = (const float*)d_in[6];
    const float* b_att    = (const float*)d_in[7];
    float*       out      = (float*)d_out;

    // Workspace carve-up (256B aligned slices)
    char* ws = (char*)d_ws;
    size_t off = 0;
    auto take = [&](size_t bytes) -> char* {
        char* p = ws + off;
        off += (bytes + 255) & ~(size_t)255;
        return p;
    };
    float* Wh      = (float*)take((size_t)N_NODES * OUT_F * 4);   // 10.24 MB
    float* ssrc    = (float*)take((size_t)N_NODES * 4);
    float* sdst    = (float*)take((size_t)N_NODES * 4);
    float* colpart = (float*)take((size_t)64 * OUT_F * 4);
    float* colsum  = (float*)take((size_t)OUT_F * 4);
    int*   deg     = (int*)  take((size_t)N_NODES * 4);
    int*   rowptr  = (int*)  take((size_t)(N_NODES + 1) * 4);
    int*   cursor  = (int*)  take((size_t)N_NODES * 4);
    int*   colidx  = (int*)  take((size_t)2 * N_EDGES * 4);       // 2.56 MB
    float* wval    = (float*)take((size_t)2 * N_EDGES * 4);       // 2.56 MB
    (void)ws_size; (void)in_sizes; (void)n_in; (void)out_size;

    // 1) WMMA GEMM: Wh = features @ W + b_lin
    gat_wh_gemm<<<dim3(N_NODES / 16, 4), dim3(32, 4), 0, stream>>>(features, W, b_lin, Wh);

    // 2) per-node attention scores
    gat_scores<<<(N_NODES + 7) / 8, 256, 0, stream>>>(Wh, Wa_src, Wa_dst, ssrc, sdst);

    // 3) column sum of Wh (deterministic two-stage)
    gat_colsum_partial<<<64, 256, 0, stream>>>(Wh, colpart);
    gat_colsum_final<<<1, 256, 0, stream>>>(colpart, colsum);

    // 4) CSR build
    hipMemsetAsync(deg, 0, (size_t)N_NODES * 4, stream);
    gat_degree<<<(N_EDGES + 255) / 256, 256, 0, stream>>>(edge_u, edge_v, deg);
    gat_scan<<<1, 1024, 0, stream>>>(deg, rowptr);
    hipMemcpyAsync(cursor, rowptr, (size_t)N_NODES * 4, hipMemcpyDeviceToDevice, stream);
    gat_fill<<<(N_EDGES + 255) / 256, 256, 0, stream>>>(edge_u, edge_v, ssrc, sdst, b_att,
                                                        cursor, colidx, wval);

    // 5) sparse aggregation + softmax-normalize + ELU
    gat_aggregate<<<N_NODES, 256, 0, stream>>>(Wh, colsum, rowptr, colidx, wval, out);
}